// ALNNLayer_44470091382993
// MI455X (gfx1250) — compile-verified
//
#include <hip/hip_runtime.h>
#include <cstdint>

#define B_ 32
#define L_ 128
#define D_ 48
#define R_ 64

typedef float v2f __attribute__((ext_vector_type(2)));
typedef float v8f __attribute__((ext_vector_type(8)));

__device__ __forceinline__ uint32_t lds_addr_of(const void* p) {
  // For LDS (shared) pointers the low 32 bits of the flat address are the
  // wave-relative LDS byte offset (CDNA5 aperture mapping).
  return (uint32_t)(uintptr_t)p;
}

// gfx1250 async copy global -> LDS (GVS mode: saddr base + 32-bit vgpr offset).
__device__ __forceinline__ void async_load_b128(uint32_t lds, const void* base, uint32_t goff) {
  asm volatile("global_load_async_to_lds_b128 %0, %1, %2"
               :: "v"(lds), "v"(goff), "s"(base)
               : "memory");
}

__global__ __launch_bounds__(256, 2)
void alnn_fused_kernel(const float* __restrict__ X,  const float* __restrict__ T,
                       const float* __restrict__ M,  const float* __restrict__ DT,
                       const float* __restrict__ P,  const float* __restrict__ alpha,
                       const float* __restrict__ w_t, const float* __restrict__ b_t,
                       const float* __restrict__ w_v, const float* __restrict__ b_v,
                       float* __restrict__ out)
{
  __shared__ __align__(16) float s_wt[L_ * 16 * 5];  // 40 KB: w_t[r, :, d0:d0+16, :]
  __shared__ __align__(16) float s_bt[L_ * 16];      //  8 KB
  __shared__ __align__(16) float s_wv[L_ * 16];      //  8 KB
  __shared__ __align__(16) float s_part[16 * 16];    //  1 KB: partials (b_loc, lq, d_loc)

  const int bid    = blockIdx.x;
  const int r      = bid / 24;       // 64 r values
  const int rem    = bid % 24;
  const int dtile  = rem / 8;        // 3 tiles of 16 d
  const int bgrp   = rem % 8;        // 8 groups of 4 b
  const int d0     = dtile * 16;
  const int b_base = bgrp * 4;

  const int t = threadIdx.x;

  // ---- stage r-sliced weights into LDS via gfx1250 async-tensor path ----
  {
    const float* wt_base = w_t + (size_t)r * L_ * D_ * 5;
    // per l-row: 16 d * 5 k = 80 contiguous floats = 20 x b128 chunks
    for (int c = t; c < L_ * 20; c += 256) {               // 2560 chunks, 10/thread
      const int l = c / 20, cc = c % 20;
      const uint32_t goff = (uint32_t)((l * D_ * 5 + d0 * 5) * 4 + cc * 16);
      async_load_b128(lds_addr_of(&s_wt[l * 80 + cc * 4]), wt_base, goff);
    }
    const float* bt_base = b_t + (size_t)r * L_ * D_;
    const float* wv_base = w_v + (size_t)r * L_ * D_;
    for (int c = t; c < L_ * 4; c += 256) {                // 512 chunks each
      const int l = c / 4, cc = c % 4;
      const uint32_t goff = (uint32_t)((l * D_ + d0) * 4 + cc * 16);
      async_load_b128(lds_addr_of(&s_bt[l * 16 + cc * 4]), bt_base, goff);
      async_load_b128(lds_addr_of(&s_wv[l * 16 + cc * 4]), wv_base, goff);
    }
  }
  asm volatile("s_wait_asynccnt 0" ::: "memory");
  __syncthreads();

  // ---- thread mapping: d_loc (16) x l-quarter (4) x b_loc (4) ----
  const int d_loc = t & 15;
  const int lq    = (t >> 4) & 3;
  const int b_loc = t >> 6;
  const int b     = b_base + b_loc;
  const int d     = d0 + d_loc;

  const float a_r   = fmaxf(alpha[r], 0.0f);
  const float ref_r = 48.0f * (float)r / 63.0f;   // linspace(0, 48, 64)

  float acc = 0.0f;
  const int l0 = lq * 32;
  #pragma unroll 4
  for (int i = 0; i < 32; ++i) {
    const int l  = l0 + i;
    const int gi = (b * L_ + l) * D_ + d;
    const float x  = X[gi];
    const float tt = T[gi];
    const float m  = M[gi];
    const float dt = DT[gi];
    const float p  = P[gi];

    const float dist  = fabsf(tt - ref_r);
    const float kern  = __expf(-a_r * dist);
    const float inten = fmaxf(x * kern, 0.0f);

    const int wi = l * 16 + d_loc;
    const float* wt = &s_wt[wi * 5];
    float s = wt[0] * x;
    s = fmaf(wt[1], inten, s);
    s = fmaf(wt[2], m,     s);
    s = fmaf(wt[3], dt,    s);
    s = fmaf(wt[4], p,     s);
    s = fmaf(5.0f, s_bt[wi], s);          // b_t broadcast over the 5-axis sum
    const float h = fmaxf(s, 0.0f);
    acc = fmaf(s_wv[wi], h, acc);
  }

  s_part[(b_loc * 4 + lq) * 16 + d_loc] = acc;
  __syncthreads();

  // ---- final reduction over the 4 l-quarters with v_wmma_f32_16x16x4_f32 ----
  // A[m=d, k=lq] = partial; B = all-ones 4x16  =>  D[m,n] = Sum_k A[m,k] (all n equal)
  const int wid  = t >> 5;   // wave id (uniform per wave)
  const int lane = t & 31;
  if (wid < 4) {             // wave-uniform branch: EXEC all-ones inside
    const int bb  = wid;                 // one wave per b in the group
    const int row = lane & 15;           // M index
    const int sel = lane >> 4;           // lanes 0-15: K=0,1 ; lanes 16-31: K=2,3
    v2f A;
    A.x = s_part[(bb * 4 + 2 * sel + 0) * 16 + row];
    A.y = s_part[(bb * 4 + 2 * sel + 1) * 16 + row];
    v2f Bones; Bones.x = 1.0f; Bones.y = 1.0f;
    v8f C = {};
    C = __builtin_amdgcn_wmma_f32_16x16x4_f32(
        /*neg_a=*/false, A, /*neg_b=*/false, Bones,
        /*c_mod=*/(short)0, C, /*reuse_a=*/false, /*reuse_b=*/false);

    // D layout: lanes 0-15 VGPR j -> M=j ; lanes 16-31 VGPR j -> M=j+8 (all N equal)
    if ((lane & 15) == 0) {
      const int dbase = (lane >> 4) * 8;
      const int bg    = b_base + bb;
      #pragma unroll
      for (int j = 0; j < 8; ++j) {
        const int dd = d0 + dbase + j;
        const float bias = 128.0f * b_v[r * D_ + dd];    // b_v broadcast over L
        const float v = fmaxf(C[j] + bias, 0.0f);
        out[(bg * R_ + r) * D_ + dd] = v;
      }
    }
  }
}

extern "C" void kernel_launch(void* const* d_in, const int* in_sizes, int n_in,
                              void* d_out, int out_size, void* d_ws, size_t ws_size,
                              hipStream_t stream) {
  (void)in_sizes; (void)n_in; (void)out_size; (void)d_ws; (void)ws_size;
  const float* X     = (const float*)d_in[0];
  const float* T     = (const float*)d_in[1];
  const float* M     = (const float*)d_in[2];
  const float* DT    = (const float*)d_in[3];
  const float* P     = (const float*)d_in[4];
  const float* alpha = (const float*)d_in[5];
  const float* w_t   = (const float*)d_in[6];
  const float* b_t   = (const float*)d_in[7];
  const float* w_v   = (const float*)d_in[8];
  const float* b_v   = (const float*)d_in[9];
  float* out = (float*)d_out;

  dim3 grid(R_ * 3 * (B_ / 4));   // 64 r * 3 d-tiles * 8 b-groups = 1536 blocks
  dim3 block(256);                // 8 wave32
  alnn_fused_kernel<<<grid, block, 0, stream>>>(X, T, M, DT, P, alpha,
                                                w_t, b_t, w_v, b_v, out);
}